// ContextInnerProd_25134148616293
// MI455X (gfx1250) — compile-verified
//
#include <hip/hip_runtime.h>

typedef __attribute__((ext_vector_type(2))) float v2f;
typedef __attribute__((ext_vector_type(8))) float v8f;

#define SEQ_LEN 16384
#define ENC_H   2048
#define DEC_H   2048
#define KSPLIT  4            // K-dim split for the scores pass (occupancy)
#define KCHUNK  (ENC_H / KSPLIT)       // 512
#define S_TILES 128          // context partial tiles (16384/128 = 128 rows each)
#define ROWS_PER_STILE (SEQ_LEN / S_TILES)

// ---------------- Workspace layout (in floats) ----------------
#define WS_V       0                           // 2048
#define WS_SPART   2048                        // KSPLIT * 16384 = 65536
#define WS_SCORES  (2048 + KSPLIT * SEQ_LEN)   // 67584 .. +16384
#define WS_STATS   (WS_SCORES + SEQ_LEN)       // 83968 .. +2
#define WS_PARTIAL 83984                       // 16-float aligned; S_TILES*2048

// ---------------------------------------------------------------
// Kernel 1: v[row] = W[row,:] . d   (2048 blocks x 256 threads)
// ---------------------------------------------------------------
__global__ void k_matvec_v(const float* __restrict__ W,
                           const float* __restrict__ d,
                           float* __restrict__ v) {
    __shared__ float red[8];
    const int row = blockIdx.x;
    const float* wr = W + (size_t)row * DEC_H;
    float s = 0.f;
    for (int i = threadIdx.x * 4; i < DEC_H; i += 256 * 4) {
        float4 a = *(const float4*)(wr + i);
        float4 b = *(const float4*)(d + i);
        s += a.x * b.x + a.y * b.y + a.z * b.z + a.w * b.w;
    }
    for (int o = 16; o > 0; o >>= 1) s += __shfl_down(s, o, 32);
    if ((threadIdx.x & 31) == 0) red[threadIdx.x >> 5] = s;
    __syncthreads();
    if (threadIdx.x == 0) {
        float t = 0.f;
        #pragma unroll
        for (int i = 0; i < 8; ++i) t += red[i];
        v[row] = t;
    }
}

// ---------------------------------------------------------------
// Kernel 2: partial scores via V_WMMA_F32_16X16X4_F32, K split 4 ways.
// Grid: (SEQ_LEN/128, KSPLIT); block = 256 threads = 8 waves.
// One wave per 16-row tile over a 512-wide K chunk -> 4096 waves total
// (enough in-flight b64 traffic to cover the HBM bandwidth-delay product).
// A (16x4 f32) ISA layout: VGPR0 = K{0|2}, VGPR1 = K{1|3}; lanes 0-15 at
// K base 0, lanes 16-31 at K base 2 -> exactly one b64 load per WMMA.
// B (4x16 f32): v[k..k+3] broadcast across all 16 columns (same per-lane
// b64 pattern, sourced from LDS); every column of D = the 16 row scores.
// ---------------------------------------------------------------
__global__ void k_scores_wmma(const float* __restrict__ E,
                              const float* __restrict__ v,
                              float* __restrict__ spart) {
    __shared__ float vs[KCHUNK];
    const int k0 = blockIdx.y * KCHUNK;
    for (int i = threadIdx.x; i < KCHUNK; i += blockDim.x) vs[i] = v[k0 + i];
    __syncthreads();

    const int wave    = threadIdx.x >> 5;          // 0..7
    const int lane    = threadIdx.x & 31;
    const int rowBase = blockIdx.x * 128 + wave * 16;
    const int m       = lane & 15;
    const int kHalf   = (lane >> 4) * 2;           // 0 or 2
    const float* rowPtr = E + (size_t)(rowBase + m) * ENC_H + k0 + kHalf;

    v8f c = {};
    for (int k = 0; k < KCHUNK; k += 4) {
        v2f a = *(const v2f*)(rowPtr + k);         // A: E tile 16x4
        v2f b = *(const v2f*)(&vs[k + kHalf]);     // B: v chunk bcast over N
        c = __builtin_amdgcn_wmma_f32_16x16x4_f32(
                /*neg_a=*/false, a, /*neg_b=*/false, b,
                /*c_mod=*/(short)0, c, /*reuse_a=*/false, /*reuse_b=*/false);
    }
    // D layout: VGPR r -> M=r (lanes 0-15) / M=8+r (lanes 16-31), same for
    // every N column. Lanes 0 and 16 emit the 16 partial scores.
    if (lane == 0 || lane == 16) {
        const int r0 = rowBase + ((lane >> 4) * 8);
        float* dst = spart + (size_t)blockIdx.y * SEQ_LEN + r0;
        #pragma unroll
        for (int r = 0; r < 8; ++r) dst[r] = c[r];
    }
}

// ---------------------------------------------------------------
// Kernel 2b: scores = sum of KSPLIT partials (deterministic order)
// ---------------------------------------------------------------
__global__ void k_score_sum(const float* __restrict__ spart,
                            float* __restrict__ scores) {
    const int e = blockIdx.x * 256 + threadIdx.x;
    float s = 0.f;
    #pragma unroll
    for (int c = 0; c < KSPLIT; ++c) s += spart[(size_t)c * SEQ_LEN + e];
    scores[e] = s;
}

// ---------------------------------------------------------------
// Kernel 3: stable-softmax stats over 16384 scores (1 block, 1024 thr)
// ---------------------------------------------------------------
__global__ void k_softmax_stats(const float* __restrict__ scores,
                                float* __restrict__ stats) {
    __shared__ float red[33];
    float m = -3.402823466e38f;
    for (int i = threadIdx.x; i < SEQ_LEN; i += 1024) m = fmaxf(m, scores[i]);
    for (int o = 16; o > 0; o >>= 1) m = fmaxf(m, __shfl_down(m, o, 32));
    if ((threadIdx.x & 31) == 0) red[threadIdx.x >> 5] = m;
    __syncthreads();
    if (threadIdx.x == 0) {
        float t = red[0];
        for (int w = 1; w < 32; ++w) t = fmaxf(t, red[w]);
        red[32] = t;
    }
    __syncthreads();
    const float gmax = red[32];
    __syncthreads();

    float s = 0.f;
    for (int i = threadIdx.x; i < SEQ_LEN; i += 1024) s += __expf(scores[i] - gmax);
    for (int o = 16; o > 0; o >>= 1) s += __shfl_down(s, o, 32);
    if ((threadIdx.x & 31) == 0) red[threadIdx.x >> 5] = s;
    __syncthreads();
    if (threadIdx.x == 0) {
        float t = 0.f;
        for (int w = 0; w < 32; ++w) t += red[w];
        stats[0] = gmax;
        stats[1] = t;
    }
}

// ---------------------------------------------------------------
// Kernel 4: partial context. S_TILES blocks, each handles 128 rows of E
// across all 2048 columns (8 cols/thread in registers, float4 loads).
// Second pass over E should be L2-resident (134 MB < 192 MB L2).
// Deterministic: partials written per tile, no FP atomics.
// ---------------------------------------------------------------
__global__ void k_context_partial(const float* __restrict__ E,
                                  const float* __restrict__ scores,
                                  const float* __restrict__ stats,
                                  float* __restrict__ partial) {
    const int tile = blockIdx.x;             // 0..S_TILES-1
    const int s0   = tile * ROWS_PER_STILE;
    const float gmax = stats[0];
    const float inv  = 1.0f / stats[1];
    const int col = threadIdx.x * 8;         // 256 threads * 8 = 2048 cols

    float acc[8] = {0.f, 0.f, 0.f, 0.f, 0.f, 0.f, 0.f, 0.f};
    for (int s = 0; s < ROWS_PER_STILE; ++s) {
        const float w = __expf(scores[s0 + s] - gmax) * inv;
        const float4* p = (const float4*)(E + (size_t)(s0 + s) * ENC_H + col);
        const float4 e0 = p[0];
        const float4 e1 = p[1];
        acc[0] += w * e0.x; acc[1] += w * e0.y;
        acc[2] += w * e0.z; acc[3] += w * e0.w;
        acc[4] += w * e1.x; acc[5] += w * e1.y;
        acc[6] += w * e1.z; acc[7] += w * e1.w;
    }
    float4* out = (float4*)(partial + (size_t)tile * ENC_H + col);
    out[0] = make_float4(acc[0], acc[1], acc[2], acc[3]);
    out[1] = make_float4(acc[4], acc[5], acc[6], acc[7]);
}

// ---------------------------------------------------------------
// Kernel 5: reduce S_TILES partials -> context[2048]  (8 blocks x 256 thr)
// ---------------------------------------------------------------
__global__ void k_context_reduce(const float* __restrict__ partial,
                                 float* __restrict__ out) {
    const int e = blockIdx.x * 256 + threadIdx.x;
    float s = 0.f;
    for (int t = 0; t < S_TILES; ++t) s += partial[(size_t)t * ENC_H + e];
    out[e] = s;
}

// ---------------------------------------------------------------
extern "C" void kernel_launch(void* const* d_in, const int* in_sizes, int n_in,
                              void* d_out, int out_size, void* d_ws, size_t ws_size,
                              hipStream_t stream) {
    const float* E = (const float*)d_in[0];   // [16384, 2048]
    const float* d = (const float*)d_in[1];   // [1, 2048]
    const float* W = (const float*)d_in[2];   // [2048, 2048]
    float* out = (float*)d_out;               // [1, 2048]
    float* ws  = (float*)d_ws;

    float* v       = ws + WS_V;
    float* spart   = ws + WS_SPART;
    float* scores  = ws + WS_SCORES;
    float* stats   = ws + WS_STATS;
    float* partial = ws + WS_PARTIAL;

    k_matvec_v       <<<ENC_H, 256, 0, stream>>>(W, d, v);
    k_scores_wmma    <<<dim3(SEQ_LEN / 128, KSPLIT), 256, 0, stream>>>(E, v, spart);
    k_score_sum      <<<SEQ_LEN / 256, 256, 0, stream>>>(spart, scores);
    k_softmax_stats  <<<1, 1024, 0, stream>>>(scores, stats);
    k_context_partial<<<S_TILES, 256, 0, stream>>>(E, scores, stats, partial);
    k_context_reduce <<<ENC_H / 256, 256, 0, stream>>>(partial, out);
}